// logistic_fun_111669150015
// MI455X (gfx1250) — compile-verified
//
#include <hip/hip_runtime.h>
#include <hip/hip_bf16.h>

// ---------------------------------------------------------------------------
// Polynomial (order<=3, D=16) logistic regression, restructured as GEMM:
//   Y[n,q]   = Xp[n,b_q]*Xp[n,c_q]        (153 pairs over 17 dims, pad K=160)
//   Z        = Y @ W2^T                   (f32 WMMA 16x16x4, K-loop of 40)
//   logit[n] = w_const + sum_a X[n,a]*Z[n,a] ;  out = sigmoid(logit)
// Pure f32 end-to-end via V_WMMA_F32_16X16X4_F32 (matches f32 reference).
//
// v2: A operands built from register-resident Xp rows inside divergent
//     half-wave branches (compile-time pair indices -> zero DS gathers,
//     zero address registers); B served from block-shared LDS with
//     immediate-offset ds_load_b64 (1 DS op per WMMA, conflict-free).
// ---------------------------------------------------------------------------

typedef __attribute__((ext_vector_type(2))) float v2f;
typedef __attribute__((ext_vector_type(8))) float v8f;

#define DP 17            // 16 features + ones column (sentinel index 16)
#define KDIM 160         // padded K (pad pairs map to 1*1 with zero weights)
#define KSTEPS 40        // KDIM / 4 (f32 WMMA does K=4 per instruction)
#define XSTR 20          // LDS row stride in floats (80B -> 16B aligned rows)
#define BLDS_N 2560      // packed B buffer: 80 pair-rows * 16 cols * 2 floats

// Compile-time (b,c) pair table: q -> (b_q, c_q), b<=c, row-major triangle.
struct PairTab { int b[KDIM]; int c[KDIM]; };
static constexpr PairTab makeTab() {
  PairTab t{};
  int q = 0;
  for (int b = 0; b < DP; ++b)
    for (int c = b; c < DP; ++c) { t.b[q] = b; t.c[q] = c; ++q; }
  for (; q < KDIM; ++q) { t.b[q] = DP - 1; t.c[q] = DP - 1; } // pad: Y=1, W=0
  return t;
}
static constexpr PairTab TAB = makeTab();

// ---------------------------------------------------------------------------
// Prep kernel (1 block): scatter w[969] into packed B buffer in workspace.
// ws layout (floats): [0 .. 2559]  Bpk[k2][n][2] with
//     Bpk[k2][n] = { W2t[2*k2][n], W2t[2*k2+1][n] },  W2t[K][a] = W2[a][K]
//   [2560]       w_const  (order-0 monomial, a == 16)
// ---------------------------------------------------------------------------
__global__ void build_weights_kernel(const float* __restrict__ w,
                                     const int* __restrict__ comb_idx,
                                     float* __restrict__ ws) {
  const int tid = threadIdx.x;
  for (int i = tid; i < BLDS_N + 1; i += 256) ws[i] = 0.0f;
  __syncthreads();
  for (int i = tid; i < 969; i += 256) {
    const int a = comb_idx[3 * i + 0];
    const int b = comb_idx[3 * i + 1];
    const int c = comb_idx[3 * i + 2];
    const float wi = w[i];
    if (a == 16) {                 // constant monomial (only i == 0)
      ws[BLDS_N] = wi;
    } else {
      // triangular pair index: q = offset(b) + (c - b), offset(b)=17b-b(b-1)/2
      const int q = b * DP - (b * (b - 1)) / 2 + (c - b);
      const int pos = ((q >> 1) * 16 + a) * 2 + (q & 1);
      ws[pos] = wi;                // injective scatter (sorted monomials)
    }
  }
}

// ---------------------------------------------------------------------------
// Main kernel: 128 blocks x 256 threads = 1024 waves; 4 tiles of 16 samples
// per wave = 4096 tiles = 65536 samples.
// ---------------------------------------------------------------------------
__global__ __launch_bounds__(256) void poly_logistic_kernel(
    const float* __restrict__ X, const float* __restrict__ ws,
    float* __restrict__ out, int tiles_per_wave) {
  __shared__ __align__(16) float Blds[BLDS_N];      // packed W2^T, block-wide
  __shared__ __align__(16) float xp[8][16 * XSTR];  // per-wave Xp tile

  const int tid  = threadIdx.x;
  const int wave = tid >> 5;
  const int lane = tid & 31;
  const int hi   = lane >> 4;   // 0: K-phase {0,1};  1: K-phase {2,3}
  const int lm   = lane & 15;   // sample row (A side) / output col (B,C side)
  float* xw = &xp[wave][0];

  // Stage packed B into LDS once per block (shared by all 8 waves).
  for (int i = tid; i < BLDS_N; i += 256) Blds[i] = ws[i];
  const float wconst = ws[BLDS_N];

  const int wave_global = blockIdx.x * 8 + wave;
  const int tile0 = wave_global * tiles_per_wave;

  // Per-lane B base: element (k2*16+lm)*2 floats, k2 = 2k+hi
  //   -> float offset 64k + (hi*16+lm)*2 ; lanes cover all 64 banks exactly.
  const float* bp = Blds + (hi * 16 + lm) * 2;

  __syncthreads();  // Blds visible to all waves

  for (int t = 0; t < tiles_per_wave; ++t) {
    const int tile = tile0 + t;

    // --- Stage X tile (16 samples x 16 feats) to LDS (transpose step) ---
    const float4* Xv = (const float4*)(X + (size_t)tile * 256);
    const float4 v0 = Xv[lane];        // flat floats 4*lane    .. +3
    const float4 v1 = Xv[lane + 32];   // flat floats 128+4*lane.. +3
    {
      const int m0 = lane >> 2;
      const int f0 = (lane & 3) * 4;
      *(float4*)(xw + m0 * XSTR + f0)       = v0;   // rows 0..7
      *(float4*)(xw + (m0 + 8) * XSTR + f0) = v1;   // rows 8..15
    }
    // speculative prefetch of next tile's X (gfx1250 global_prefetch_b8)
    __builtin_prefetch(X + (size_t)(tile + 1) * 256 + lane * 8, 0, 0);
    __syncthreads();

    // --- Pull this lane's sample row into registers (static indexing) ---
    const float* xr = xw + lm * XSTR;
    float xv[DP];
    {
      const float4 r0 = *(const float4*)(xr + 0);
      const float4 r1 = *(const float4*)(xr + 4);
      const float4 r2 = *(const float4*)(xr + 8);
      const float4 r3 = *(const float4*)(xr + 12);
      xv[0]  = r0.x; xv[1]  = r0.y; xv[2]  = r0.z; xv[3]  = r0.w;
      xv[4]  = r1.x; xv[5]  = r1.y; xv[6]  = r1.z; xv[7]  = r1.w;
      xv[8]  = r2.x; xv[9]  = r2.y; xv[10] = r2.z; xv[11] = r2.w;
      xv[12] = r3.x; xv[13] = r3.y; xv[14] = r3.z; xv[15] = r3.w;
      xv[16] = 1.0f;                 // sentinel ones column
    }

    // --- Build all 40 A operands with compile-time pair indices.
    // Lane half 'hi' owns K phases {2hi, 2hi+1}; divergence is legal here
    // (no WMMA inside), branches reconverge to EXEC=all-1 before the MMAs.
    v2f A[KSTEPS];
    if (hi) {
#pragma unroll
      for (int k = 0; k < KSTEPS; ++k) {
        A[k].x = xv[TAB.b[4 * k + 2]] * xv[TAB.c[4 * k + 2]];
        A[k].y = xv[TAB.b[4 * k + 3]] * xv[TAB.c[4 * k + 3]];
      }
    } else {
#pragma unroll
      for (int k = 0; k < KSTEPS; ++k) {
        A[k].x = xv[TAB.b[4 * k + 0]] * xv[TAB.c[4 * k + 0]];
        A[k].y = xv[TAB.b[4 * k + 1]] * xv[TAB.c[4 * k + 1]];
      }
    }

    // --- K-loop: one conflict-free ds_load_b64 (imm offset) + one WMMA ---
    v8f acc = {};
#pragma unroll
    for (int k = 0; k < KSTEPS; ++k) {
      const v2f Bk = *(const v2f*)(bp + 64 * k);
      acc = __builtin_amdgcn_wmma_f32_16x16x4_f32(
          false, A[k], false, Bk, (short)0, acc, false, false);
    }

    // --- Epilogue: logit[m] = wconst + sum_a X[m][a] * Z[m][a] ---
    // C layout: VGPR r holds (M=r, N=lm) for lanes 0-15, (M=r+8, N=lm) hi.
#pragma unroll
    for (int r = 0; r < 8; ++r) {
      const int m = r + 8 * hi;
      float v = acc[r] * xw[m * XSTR + lm];   // Z[m][a]*X[m][a], a = lm
      v += __shfl_xor(v, 1, 32);              // reduce over the 16 'a' lanes
      v += __shfl_xor(v, 2, 32);
      v += __shfl_xor(v, 4, 32);
      v += __shfl_xor(v, 8, 32);
      if (lm == 0) {
        const float logit = v + wconst;
        out[(size_t)tile * 16 + m] = 1.0f / (1.0f + __expf(-logit));
      }
    }
    __syncthreads();  // protect Xp before next tile's stores
  }
}

extern "C" void kernel_launch(void* const* d_in, const int* in_sizes, int n_in,
                              void* d_out, int out_size, void* d_ws,
                              size_t ws_size, hipStream_t stream) {
  const float* X        = (const float*)d_in[0];  // [65536, 16] f32
  const float* w        = (const float*)d_in[1];  // [969] f32
  const int*   comb_idx = (const int*)d_in[2];    // [969, 3] i32
  float*       out      = (float*)d_out;          // [65536] f32
  float*       ws       = (float*)d_ws;           // >= 2561 floats

  build_weights_kernel<<<1, 256, 0, stream>>>(w, comb_idx, ws);

  // 4096 tiles of 16 samples; 128 blocks * 8 waves * 4 tiles = 4096 exactly.
  const int tiles_per_wave = 4;
  poly_logistic_kernel<<<128, 256, 0, stream>>>(X, ws, out, tiles_per_wave);
}